// TopKHorizonAccuracy_39187281609292
// MI455X (gfx1250) — compile-verified
//
#include <hip/hip_runtime.h>
#include <stdint.h>
#include <stddef.h>

// ---------------------------------------------------------------------------
// TopK (K=5) retrieval accuracy on MI455X (gfx1250).
//   similarity = P (4096x256 f32) x Lab^T (100000x256 f32); correct[b] iff
//   rank of true label within row b is < 5.
// bf16 WMMA GEMM streamed in 16x16 tiles + per-row count of entries beating
// threshold s_t = sim(b, true_label[b]).  Label tiles staged into LDS by the
// Tensor Data Mover (tensor_load_to_lds, TENSORcnt); rank counting via wave32
// ballots -> uniform SGPR counters (s_bcnt).  The 8-WMMA K-loop is emitted as
// one inline-asm block so every A/B fragment operand is explicitly bound.
// ---------------------------------------------------------------------------

typedef __bf16 bf16;
typedef __bf16        v8bf  __attribute__((ext_vector_type(8)));
typedef __bf16        v16bf __attribute__((ext_vector_type(16)));
typedef float         v8f   __attribute__((ext_vector_type(8)));
typedef unsigned int  u32x4 __attribute__((ext_vector_type(4)));
typedef unsigned int  u32x8 __attribute__((ext_vector_type(8)));
typedef int           i32x4 __attribute__((ext_vector_type(4)));
typedef int           i32x8 __attribute__((ext_vector_type(8)));
typedef float         f32x4 __attribute__((ext_vector_type(4)));

#define TOPK        5
#define DIM         256     // embedding dim (8 K-chunks of 32 for bf16 WMMA)
#define ROWS_PER_WG 128     // 8 waves x 16 rows -> amortize label traffic
#define WG_THREADS  256
#define LSPLIT      8       // label-space split across gridDim.y
#define TILE_BYTES  (16 * DIM * (int)sizeof(bf16))   // 8 KB per label tile

#if __has_builtin(__builtin_amdgcn_tensor_load_to_lds)
#define USE_TDM 1
#else
#define USE_TDM 0
#endif

#if __has_builtin(__builtin_amdgcn_ballot_w32)
#define HAVE_BALLOT 1
#else
#define HAVE_BALLOT 0
#endif

// Assemble one 16-element bf16 WMMA operand fragment from the two 16-byte
// chunks a lane owns (ISA 16-bit operand layout: within a 32-wide K step the
// lane carries K in [8h..8h+8) and [16+8h..16+8h+8), h = lane>=16), returned
// bit-cast to u32x8 for use as an inline-asm operand.
__device__ __forceinline__ u32x8 load_frag(const bf16* p, int half) {
  v8bf lo = *(const v8bf*)(p + 8 * half);
  v8bf hi = *(const v8bf*)(p + 16 + 8 * half);
  v16bf v = __builtin_shufflevector(lo, hi, 0, 1, 2, 3, 4, 5, 6, 7, 8, 9, 10,
                                    11, 12, 13, 14, 15);
  return __builtin_bit_cast(u32x8, v);
}

// ------------------------- f32 -> bf16 convert (8 elems/thread) -------------
__global__ __launch_bounds__(256) void cvt_bf16_kernel(
    const float* __restrict__ src, bf16* __restrict__ dst, int n8) {
  int i = blockIdx.x * blockDim.x + threadIdx.x;
  if (i >= n8) return;
  const f32x4* s4 = (const f32x4*)src;
  f32x4 a = s4[2 * i], b = s4[2 * i + 1];
  v8bf p;
  p[0] = (bf16)a.x; p[1] = (bf16)a.y; p[2] = (bf16)a.z; p[3] = (bf16)a.w;
  p[4] = (bf16)b.x; p[5] = (bf16)b.y; p[6] = (bf16)b.z; p[7] = (bf16)b.w;
  ((v8bf*)dst)[i] = p;
}

// ------------------- per-row threshold: s_t = <P[b], Lab[t_b]> --------------
__global__ __launch_bounds__(256) void thr_kernel(
    const bf16* __restrict__ Pbf, const bf16* __restrict__ Lbf,
    const int* __restrict__ labels, float* __restrict__ thr, int B, int D) {
  int b = blockIdx.x * blockDim.x + threadIdx.x;
  if (b >= B) return;
  int t = labels[b];
  const bf16* p = Pbf + (size_t)b * D;
  const bf16* l = Lbf + (size_t)t * D;
  float s = 0.f;
  for (int k = 0; k < D; ++k) s += (float)p[k] * (float)l[k];
  thr[b] = s;
}

#if USE_TDM
// Issue one TDM DMA: 2D tile of 16 rows x 256 bf16 (row stride 256) from
// global `gptr` into LDS byte offset `lds_off`.  D# per CDNA5 ISA ch.8.
__device__ __forceinline__ void tdm_load_tile(unsigned lds_off,
                                              const bf16* gptr) {
  unsigned long long ga = (unsigned long long)(uintptr_t)gptr;
  u32x4 g0;
  g0[0] = 1u;                                   // count=1, user descriptor
  g0[1] = lds_off;                              // lds_addr (bytes)
  g0[2] = (unsigned)ga;                         // global_addr[31:0]
  g0[3] = (unsigned)((ga >> 32) & 0x01FFFFFFull) | (2u << 30);  // [56:32]|type=2
  i32x8 g1;
  g1[0] = 0x00010000;         // workgroup_mask=0, data_size=1 (2 bytes)
  g1[1] = (int)(256u << 16);  // tensor_dim0 = 256 (low16)
  g1[2] = (int)(16u << 16);   // tensor_dim0 hi=0 | tensor_dim1 = 16 (low16)
  g1[3] = (int)(256u << 16);  // tensor_dim1 hi=0 | tile_dim0 = 256
  g1[4] = 16;                 // tile_dim1 = 16, tile_dim2 = 0
  g1[5] = 256;                // tensor_dim0_stride = 256 (low32)
  g1[6] = 0;
  g1[7] = 0;
  i32x4 gz = {0, 0, 0, 0};
#if defined(__clang_major__) && __clang_major__ >= 23
  i32x8 gz8 = {0, 0, 0, 0, 0, 0, 0, 0};
  __builtin_amdgcn_tensor_load_to_lds(g0, g1, gz, gz, gz8, 0);
#else
  __builtin_amdgcn_tensor_load_to_lds(g0, g1, gz, gz, 0);
#endif
}
#endif  // USE_TDM

// --------------------- main: bf16 WMMA GEMM + rank counting -----------------
__global__ __launch_bounds__(WG_THREADS) void simcount_kernel(
    const bf16* __restrict__ Pbf, const bf16* __restrict__ Lbf,
    const float* __restrict__ thr, const int* __restrict__ labels,
    int* __restrict__ cnt, int ntiles) {
  // double-buffered label tile: 2 x (16 labels x 256 bf16) = 16 KB LDS.
  // Sole __shared__ object -> LDS byte offset 0 (TDM lds_addr relies on this:
  // buffer i lives at i*TILE_BYTES).
  __shared__ __align__(16) bf16 sB[2][16 * DIM];

  const int tid  = threadIdx.x;
  const int wave = tid >> 5;
  const int lane = tid & 31;
  const int half = lane >> 4;   // 0: lanes 0-15, 1: lanes 16-31
  const int ln16 = lane & 15;
  const int rowbase = blockIdx.x * ROWS_PER_WG + wave * 16;

  // A fragments resident in registers: 16 rows x 256 K (64 VGPRs).
  u32x8 A[8];
  {
    const bf16* ap = Pbf + (size_t)(rowbase + ln16) * DIM;
#pragma unroll
    for (int c = 0; c < 8; ++c) A[c] = load_frag(ap + c * 32, half);
  }

  // per-lane thresholds / true labels for the 8 C-rows this lane produces
  float thrv[8];
  int   tbv[8];
#pragma unroll
  for (int r = 0; r < 8; ++r) {
    int b   = rowbase + half * 8 + r;
    thrv[r] = thr[b];
    tbv[r]  = labels[b];
  }

#if HAVE_BALLOT
  // Uniform (SGPR) counters: ballot bit i belongs to lane i, so for VGPR row
  // r the low 16 bits count row (rowbase+r), high 16 bits row (rowbase+8+r).
  int cLo[8] = {0, 0, 0, 0, 0, 0, 0, 0};
  int cHi[8] = {0, 0, 0, 0, 0, 0, 0, 0};
#else
  int cntr[8] = {0, 0, 0, 0, 0, 0, 0, 0};
#endif

  // 16x16 tile: all 8 bf16 WMMAs emitted in one asm block with explicitly
  // bound operands (two D->C accumulation chains; D->C needs no NOPs, the
  // trailing 4 v_nops cover the WMMA->VALU read of the accumulators).
  auto compute_tile = [&](const bf16* bp0, int t) {
    const bf16* bp = bp0 + ln16 * DIM;  // lane holds label column N = lane%16
    u32x8 B0 = load_frag(bp + 0 * 32, half);
    u32x8 B1 = load_frag(bp + 1 * 32, half);
    u32x8 B2 = load_frag(bp + 2 * 32, half);
    u32x8 B3 = load_frag(bp + 3 * 32, half);
    u32x8 B4 = load_frag(bp + 4 * 32, half);
    u32x8 B5 = load_frag(bp + 5 * 32, half);
    u32x8 B6 = load_frag(bp + 6 * 32, half);
    u32x8 B7 = load_frag(bp + 7 * 32, half);
    v8f acc0, acc1;
    asm("v_wmma_f32_16x16x32_bf16 %[d0], %[a0], %[b0], 0\n\t"
        "v_wmma_f32_16x16x32_bf16 %[d1], %[a1], %[b1], 0\n\t"
        "v_wmma_f32_16x16x32_bf16 %[d0], %[a2], %[b2], %[d0]\n\t"
        "v_wmma_f32_16x16x32_bf16 %[d1], %[a3], %[b3], %[d1]\n\t"
        "v_wmma_f32_16x16x32_bf16 %[d0], %[a4], %[b4], %[d0]\n\t"
        "v_wmma_f32_16x16x32_bf16 %[d1], %[a5], %[b5], %[d1]\n\t"
        "v_wmma_f32_16x16x32_bf16 %[d0], %[a6], %[b6], %[d0]\n\t"
        "v_wmma_f32_16x16x32_bf16 %[d1], %[a7], %[b7], %[d1]\n\t"
        "v_nop\n\t"
        "v_nop\n\t"
        "v_nop\n\t"
        "v_nop"
        : [d0] "=&v"(acc0), [d1] "=&v"(acc1)
        : [a0] "v"(A[0]), [a1] "v"(A[1]), [a2] "v"(A[2]), [a3] "v"(A[3]),
          [a4] "v"(A[4]), [a5] "v"(A[5]), [a6] "v"(A[6]), [a7] "v"(A[7]),
          [b0] "v"(B0), [b1] "v"(B1), [b2] "v"(B2), [b3] "v"(B3),
          [b4] "v"(B4), [b5] "v"(B5), [b6] "v"(B6), [b7] "v"(B7));
    v8f acc = acc0 + acc1;
    const int lcol = t * 16 + ln16;
#pragma unroll
    for (int r = 0; r < 8; ++r) {
      float v  = acc[r];
      bool hit = (v > thrv[r]) & (lcol != tbv[r]);  // exclude the label itself
#if HAVE_BALLOT
      unsigned m = (unsigned)__builtin_amdgcn_ballot_w32(hit);
      cLo[r] += __builtin_popcount(m & 0xFFFFu);
      cHi[r] += __builtin_popcount(m >> 16);
#else
      cntr[r] += hit ? 1 : 0;
#endif
    }
  };

  int t = blockIdx.y;  // tile index, strided by LSPLIT

#if USE_TDM
  // --- TDM-staged pipeline: one DMA in flight, one barrier per iteration ---
  if (wave == 0) tdm_load_tile(0u, Lbf + (size_t)t * 16 * DIM);
  int cur = 0;
  for (; t < ntiles; t += LSPLIT) {
    if (wave == 0) __builtin_amdgcn_s_wait_tensorcnt(0);  // buf[cur] landed
    __syncthreads();  // all waves: data ready, and buf[cur^1] free to refill
    const int nt = t + LSPLIT;
    if (wave == 0 && nt < ntiles)
      tdm_load_tile((unsigned)((cur ^ 1) * TILE_BYTES),
                    Lbf + (size_t)nt * 16 * DIM);
    if (t + 2 * LSPLIT < ntiles)  // warm L2 two tiles ahead
      __builtin_prefetch(Lbf + (size_t)(t + 2 * LSPLIT) * 16 * DIM + tid * 16,
                         0, 1);
    compute_tile(&sB[cur][0], t);
    cur ^= 1;
  }
#else
  // --- fallback: register-staged double buffering ---
  {
    const u32x4* g = (const u32x4*)(Lbf + (size_t)t * 16 * DIM);
    u32x4 r0 = g[2 * tid], r1 = g[2 * tid + 1];
    u32x4* s = (u32x4*)&sB[0][0];
    s[2 * tid] = r0;
    s[2 * tid + 1] = r1;
  }
  __syncthreads();
  int cur = 0;
  for (; t < ntiles; t += LSPLIT) {
    const int  nt   = t + LSPLIT;
    const bool have = nt < ntiles;
    u32x4 p0, p1;
    if (have) {
      const u32x4* g = (const u32x4*)(Lbf + (size_t)nt * 16 * DIM);
      p0 = g[2 * tid];
      p1 = g[2 * tid + 1];
    }
    if (t + 2 * LSPLIT < ntiles)
      __builtin_prefetch(Lbf + (size_t)(t + 2 * LSPLIT) * 16 * DIM + tid * 16,
                         0, 1);
    compute_tile(&sB[cur][0], t);
    __syncthreads();
    if (have) {
      u32x4* s = (u32x4*)&sB[cur ^ 1][0];
      s[2 * tid] = p0;
      s[2 * tid + 1] = p1;
    }
    __syncthreads();
    cur ^= 1;
  }
#endif

  // publish per-row counts
#if HAVE_BALLOT
  if (lane == 0) {
#pragma unroll
    for (int r = 0; r < 8; ++r) {
      atomicAdd(&cnt[rowbase + r], cLo[r]);
      atomicAdd(&cnt[rowbase + 8 + r], cHi[r]);
    }
  }
#else
#pragma unroll
  for (int r = 0; r < 8; ++r) {
    int s = cntr[r];
    s += __shfl_xor(s, 1, 16);
    s += __shfl_xor(s, 2, 16);
    s += __shfl_xor(s, 4, 16);
    s += __shfl_xor(s, 8, 16);
    if (ln16 == 0) atomicAdd(&cnt[rowbase + half * 8 + r], s);
  }
#endif
}

// ------------------------------ finalize ------------------------------------
__global__ __launch_bounds__(WG_THREADS) void finalize_kernel(
    const int* __restrict__ cnt, int B, float* __restrict__ out) {
  __shared__ int sred[WG_THREADS];
  int tid = threadIdx.x;
  int s = 0;
  for (int i = tid; i < B; i += WG_THREADS) s += (cnt[i] < TOPK) ? 1 : 0;
  sred[tid] = s;
  __syncthreads();
  for (int off = WG_THREADS / 2; off > 0; off >>= 1) {
    if (tid < off) sred[tid] += sred[tid + off];
    __syncthreads();
  }
  if (tid == 0) out[0] = (float)sred[0] / (float)B;
}

// ------------------------------ launcher ------------------------------------
extern "C" void kernel_launch(void* const* d_in, const int* in_sizes, int n_in,
                              void* d_out, int out_size, void* d_ws, size_t ws_size,
                              hipStream_t stream) {
  (void)n_in; (void)out_size; (void)ws_size;
  const float* P      = (const float*)d_in[0];  // (B, D) f32
  const float* Lab    = (const float*)d_in[1];  // (L, D) f32
  const int*   labels = (const int*)d_in[2];    // (B,)   int

  const size_t nP = (size_t)in_sizes[0];
  const size_t nL = (size_t)in_sizes[1];
  const int    B  = in_sizes[2];
  const int    D  = (int)(nP / (size_t)B);      // expected 256
  const int    L  = (int)(nL / (size_t)D);      // expected 100000
  const int    ntiles = L / 16;                 // 6250 (exact)

  // workspace layout: [P bf16][Lab bf16][thr f32][cnt i32]  (~53.4 MB)
  bf16*  Pbf = (bf16*)d_ws;
  bf16*  Lbf = (bf16*)((char*)d_ws + nP * sizeof(bf16));
  float* thr = (float*)((char*)d_ws + (nP + nL) * sizeof(bf16));
  int*   cnt = (int*)((char*)thr + (size_t)B * sizeof(float));

  const int n8p = (int)(nP / 8), n8l = (int)(nL / 8);
  cvt_bf16_kernel<<<(n8p + 255) / 256, 256, 0, stream>>>(P, Pbf, n8p);
  cvt_bf16_kernel<<<(n8l + 255) / 256, 256, 0, stream>>>(Lab, Lbf, n8l);
  thr_kernel<<<(B + 255) / 256, 256, 0, stream>>>(Pbf, Lbf, labels, thr, B, D);
  hipMemsetAsync(cnt, 0, (size_t)B * sizeof(int), stream);

  dim3 grid(B / ROWS_PER_WG, LSPLIT);
  simcount_kernel<<<grid, WG_THREADS, 0, stream>>>(Pbf, Lbf, thr, labels, cnt,
                                                   ntiles);
  finalize_kernel<<<1, WG_THREADS, 0, stream>>>(cnt, B, (float*)d_out);
}